// WeightTransformPixelBilateralFilter_75926431858853
// MI455X (gfx1250) — compile-verified
//
#include <hip/hip_runtime.h>
#include <stdint.h>

// Problem constants (match reference setup_inputs / KERNEL_SIZE / DILATION / DYNAMIC_SIZE)
#define B_      4
#define CIN     32
#define COUT    24          // DYNAMIC_SIZE: only channels 0..23 are consumed
#define NP      8           // params channels
#define H_      512
#define W_      512
#define KS      5
#define DIL     2
#define PAD     4           // (KS/2)*DIL

#define TILE    16          // 16x16 output pixels per 256-thread block (8 wave32)
#define HALO    (TILE + 2*PAD)        // 24
#define PLANE   (HALO*HALO)           // 576 floats per plane
#define NPLANES (NP + COUT)           // 32 planes staged in LDS
#define SMEM_FLOATS (NPLANES*PLANE)   // 18432 floats
#define SMEM_BYTES  (SMEM_FLOATS*4)   // 73728 B  (2-4 blocks resident per 320KB WGP)
#define NF4     (SMEM_FLOATS/4)       // 4608 float4 copies per block
#define ROWF4   (HALO/4)              // 6 float4 per halo row
#define PLANEF4 (PLANE/4)             // 144 float4 per plane

// ---- CDNA5 async global->LDS path (ASYNCcnt-tracked DMA), with portable fallback ----
#if defined(__HIP_DEVICE_COMPILE__) && defined(__gfx1250__) && \
    __has_builtin(__builtin_amdgcn_global_load_async_to_lds_b128)
#define USE_ASYNC_LDS 1
#else
#define USE_ASYNC_LDS 0
#endif

#if USE_ASYNC_LDS
typedef int v4i_ __attribute__((ext_vector_type(4)));
typedef __attribute__((address_space(1))) v4i_* gv4p;   // global
typedef __attribute__((address_space(3))) v4i_* lv4p;   // LDS

__device__ __forceinline__ void async_copy_b128(const float* src, float* dst) {
    float* s = const_cast<float*>(src);
    __builtin_amdgcn_global_load_async_to_lds_b128((gv4p)s, (lv4p)dst,
                                                   /*offset=*/0, /*cpol=*/0);
}
#endif

__global__ __launch_bounds__(256)
void WeightTransformPixelBilateralFilter_75926431858853_kernel(
        const float* __restrict__ inp,   // [B][CIN][H][W]
        const float* __restrict__ par,   // [B][NP][H][W]
        float* __restrict__ out)         // [B][COUT][H][W]
{
    extern __shared__ float smem[];      // planes: [0..7]=params, [8..31]=input ch 0..23

    const int tid = threadIdx.x;
    const int b   = blockIdx.z;
    const int gx0 = blockIdx.x * TILE;
    const int gy0 = blockIdx.y * TILE;

    // ---------------- Stage halo tiles into LDS (async DMA) ----------------
    // Addresses are clamped to valid memory; out-of-bounds taps are zero-weighted
    // later via exact coordinate tests, matching the reference's zero-padded mask.
#pragma unroll
    for (int k = 0; k < NF4 / 256; ++k) {
        const int i     = tid + k * 256;
        const int plane = i / PLANEF4;
        const int rem   = i - plane * PLANEF4;
        const int r     = rem / ROWF4;
        const int c4    = rem - r * ROWF4;

        int gy = gy0 - PAD + r;
        gy = gy < 0 ? 0 : (gy > H_ - 1 ? H_ - 1 : gy);
        int gxs = gx0 - PAD + c4 * 4;                 // 16B-aligned start
        gxs = gxs < 0 ? 0 : (gxs > W_ - 4 ? W_ - 4 : gxs);

        const float* src;
        if (plane < NP)
            src = par + (((size_t)(b * NP + plane)) * H_ + gy) * W_ + gxs;
        else
            src = inp + (((size_t)(b * CIN + (plane - NP))) * H_ + gy) * W_ + gxs;

        float* dst = &smem[plane * PLANE + r * HALO + c4 * 4];

#if USE_ASYNC_LDS
        // GLOBAL_LOAD_ASYNC_TO_LDS_B128: memory -> LDS, no VGPR data, ASYNCcnt-tracked
        async_copy_b128(src, dst);
#else
        *(float4*)dst = *(const float4*)src;
#endif
    }

#if USE_ASYNC_LDS
#if __has_builtin(__builtin_amdgcn_s_wait_asynccnt)
    __builtin_amdgcn_s_wait_asynccnt(0);
#else
    asm volatile("s_wait_asynccnt 0x0" ::: "memory");
#endif
#endif
    __syncthreads();

    // ---------------- Per-pixel bilateral filter ----------------
    const int tx = tid & (TILE - 1);
    const int ty = tid >> 4;
    const int gx = gx0 + tx;
    const int gy = gy0 + ty;

    float pc[NP];
    {
        const int coff = (ty + PAD) * HALO + (tx + PAD);
#pragma unroll
        for (int p = 0; p < NP; ++p) pc[p] = smem[p * PLANE + coff];
    }

    float acc[COUT];
#pragma unroll
    for (int c = 0; c < COUT; ++c) acc[c] = 0.0f;
    float wsum = 0.0f;

#pragma unroll
    for (int dy = 0; dy < KS; ++dy) {
#pragma unroll
        for (int dx = 0; dx < KS; ++dx) {
            const int off = (ty + dy * DIL) * HALO + (tx + dx * DIL);

            float d2 = 0.0f;
#pragma unroll
            for (int p = 0; p < NP; ++p) {
                const float d = pc[p] - smem[p * PLANE + off];
                d2 = fmaf(d, d, d2);
            }

            // exact OOB mask (reference pads with zeros => zero weight)
            const unsigned ny = (unsigned)(gy + dy * DIL - PAD);
            const unsigned nx = (unsigned)(gx + dx * DIL - PAD);
            const float w = (ny < (unsigned)H_ && nx < (unsigned)W_)
                                ? __expf(-d2) : 0.0f;
            wsum += w;

#pragma unroll
            for (int c = 0; c < COUT; ++c)
                acc[c] = fmaf(w, smem[(NP + c) * PLANE + off], acc[c]);
        }
    }

    const float inv = __builtin_amdgcn_rcpf(wsum + 1e-8f);
#pragma unroll
    for (int c = 0; c < COUT; ++c)
        out[(((size_t)(b * COUT + c)) * H_ + gy) * W_ + gx] = acc[c] * inv;
}

extern "C" void kernel_launch(void* const* d_in, const int* in_sizes, int n_in,
                              void* d_out, int out_size, void* d_ws, size_t ws_size,
                              hipStream_t stream) {
    (void)in_sizes; (void)n_in; (void)out_size; (void)d_ws; (void)ws_size;
    const float* inp = (const float*)d_in[0];   // [4][32][512][512] f32
    const float* par = (const float*)d_in[1];   // [4][8][512][512]  f32
    float* out = (float*)d_out;                 // [4][24][512][512] f32

    dim3 grid(W_ / TILE, H_ / TILE, B_);        // (32, 32, 4)
    dim3 block(256);                            // 8 wave32 per block
    WeightTransformPixelBilateralFilter_75926431858853_kernel
        <<<grid, block, SMEM_BYTES, stream>>>(inp, par, out);
}